// DifferentiablePaletteQuantization_15272903704770
// MI455X (gfx1250) — compile-verified
//
#include <hip/hip_runtime.h>

typedef float v2f __attribute__((ext_vector_type(2)));
typedef float v8f __attribute__((ext_vector_type(8)));

#if __has_builtin(__builtin_amdgcn_exp2f)
#define FAST_EXP2(x) __builtin_amdgcn_exp2f(x)
#else
#define FAST_EXP2(x) exp2f(x)
#endif

#if __has_builtin(__builtin_amdgcn_rcpf)
#define FAST_RCP(x) __builtin_amdgcn_rcpf(x)
#else
#define FAST_RCP(x) (1.0f / (x))
#endif

__device__ __forceinline__ v8f wmma_f32_16x16x4(v2f a, v2f b, v8f c) {
  // 8 args: (neg_a, A, neg_b, B, c_mod, C, reuse_a, reuse_b); neg must be 0 for f32
  return __builtin_amdgcn_wmma_f32_16x16x4_f32(false, a, false, b, (short)0, c,
                                               false, false);
}

// Problem dims: B=32, H=W=256, C=3, K=32
constexpr int BATCH = 32;
constexpr int HW    = 256 * 256;   // pixels per batch
constexpr int KPAL  = 32;
constexpr int TILE  = 16;          // pixels per WMMA tile (M)
constexpr int TPW   = 16;          // tiles per wave (amortize palette setup)
constexpr int WPB   = 8;           // waves per 256-thread block
constexpr int WAVES_PER_BATCH  = HW / (TILE * TPW);      // 256
constexpr int BLOCKS_PER_BATCH = WAVES_PER_BATCH / WPB;  // 32
constexpr int GRID = BATCH * BLOCKS_PER_BATCH;           // 1024

constexpr int ESTRIDE = 20;  // dwords per k-row of E in LDS: 16 pixels + pad,
                             // keeps b128 stores 16B-aligned, spreads banks

// 4-byte-aligned 8-byte chunk (pixel stride is 12B, so 8B loads are 4-aligned)
struct __attribute__((packed, aligned(4))) f2u {
  float x, y;
};

__global__ __launch_bounds__(256) void palette_quant_wmma(
    const float* __restrict__ img, const float* __restrict__ pal,
    const float* __restrict__ temp, float* __restrict__ out) {
  // Per-wave private LDS slices: E matrix (32 k-rows x 16 pixels) + out pack
  __shared__ float elds[WPB][KPAL * ESTRIDE];  // 8 * 2560B = 20 KB
  __shared__ float olds[WPB][TILE * 4];        // 8 *  256B =  2 KB

  const int wave = threadIdx.x >> 5;
  const int lane = threadIdx.x & 31;
  const int col  = lane & 15;   // N column of WMMA D / M row of A
  const int half = lane >> 4;   // lane-half selects K pair / M+8 rows

  const int batch = blockIdx.x / BLOCKS_PER_BATCH;
  const int wib   = (blockIdx.x % BLOCKS_PER_BATCH) * WPB + wave;
  const int pix0  = wib * (TILE * TPW);  // first pixel (within batch)

  const float* palb = pal + batch * (KPAL * 3);
  const float* imgb = img + (size_t)batch * HW * 3;
  float*       outb = out + (size_t)batch * HW * 3;

  float* myE = elds[wave];
  float* myO = olds[wave];

  // ---- per-wave setup (palette-derived constants) ----
  const float pl0 = palb[col * 3 + 0];
  const float pl1 = palb[col * 3 + 1];
  const float pl2 = palb[col * 3 + 2];
  const float ph0 = palb[(col + 16) * 3 + 0];
  const float ph1 = palb[(col + 16) * 3 + 1];
  const float ph2 = palb[(col + 16) * 3 + 2];

  const float T    = temp[0];
  const float L2E  = 1.4426950408889634f;
  const float sA   = 2.0f * L2E / T;  // cross-term scale
  const float sLo  = (pl0 * pl0 + pl1 * pl1 + pl2 * pl2) * (L2E / T);
  const float sHi  = (ph0 * ph0 + ph1 * ph1 + ph2 * ph2) * (L2E / T);

  // GEMM1 B operands: B[k=c][n=palette] = pal[n][c] (row 3 zero-padded)
  v2f b1, b2;
  b1.x = half ? pl2 : pl0;  b1.y = half ? 0.0f : pl1;
  b2.x = half ? ph2 : ph0;  b2.y = half ? 0.0f : ph1;

  // GEMM2 B operands: Q'[k][n] = (n<3 ? pal[k][n] : n==3 ? 1 : 0), 8 K-chunks
  float bx[8], by[8];
#pragma unroll
  for (int j = 0; j < 8; ++j) {
    const int k0 = 4 * j + 2 * half;
    bx[j] = (col < 3) ? palb[k0 * 3 + col]
                      : ((col == 3) ? 1.0f : 0.0f);
    by[j] = (col < 3) ? palb[(k0 + 1) * 3 + col]
                      : ((col == 3) ? 1.0f : 0.0f);
  }

  // Un-predicated 8B pixel load: half0 lanes read (c0,c1) at +0,
  // half1 lanes read (c1,c2) at +4. One global_load_b64, no exec juggling.
  const f2u* ip0 = (const f2u*)(imgb + (size_t)(pix0 + col) * 3 + half);

  // ---- software-pipelined main loop: one 16-pixel tile per iteration ----
  f2u vcur = *ip0;  // prologue: tile 0 pixel data in flight / consumed below

  for (int t = 0; t < TPW; ++t) {
    const int pbase = pix0 + t * TILE;

    // Issue next tile's pixel load NOW; it retires under this tile's math.
    f2u vnext;
    if (t + 1 < TPW) {
      vnext = *(const f2u*)(imgb + (size_t)(pbase + TILE + col) * 3 + half);
    } else {
      vnext.x = 0.0f; vnext.y = 0.0f;
    }

    // A operand from vcur: v0 = {K0 (c0) | K2 (c2)}, v1 = {K1 (c1) | K3 (pad)}
    v2f a;
    a.x = half ? vcur.y : vcur.x;
    a.y = half ? 0.0f   : vcur.y;

    // GEMM1: cross[m=pixel][n=palette] = img . pal (two 16-wide palette tiles)
    v8f zero = {};
    v8f cr1 = wmma_f32_16x16x4(a, b1, zero);  // palettes 0..15
    v8f cr2 = wmma_f32_16x16x4(a, b2, zero);  // palettes 16..31

    // logits -> exp2; |img|^2 cancels in softmax. Store E to LDS k-major.
    float ev1[8], ev2[8];
#pragma unroll
    for (int r = 0; r < 8; ++r) {
      ev1[r] = FAST_EXP2(fmaf(cr1[r], sA, -sLo));
      ev2[r] = FAST_EXP2(fmaf(cr2[r], sA, -sHi));
    }
    {
      float4 q0, q1, q2, q3;
      q0.x = ev1[0]; q0.y = ev1[1]; q0.z = ev1[2]; q0.w = ev1[3];
      q1.x = ev1[4]; q1.y = ev1[5]; q1.z = ev1[6]; q1.w = ev1[7];
      q2.x = ev2[0]; q2.y = ev2[1]; q2.z = ev2[2]; q2.w = ev2[3];
      q3.x = ev2[4]; q3.y = ev2[5]; q3.z = ev2[6]; q3.w = ev2[7];
      // E[k][p]: lane's 8 pixel values are contiguous -> 2x ds_store_b128 each
      *(float4*)(myE + col * ESTRIDE + 8 * half)            = q0;
      *(float4*)(myE + col * ESTRIDE + 8 * half + 4)        = q1;
      *(float4*)(myE + (col + 16) * ESTRIDE + 8 * half)     = q2;
      *(float4*)(myE + (col + 16) * ESTRIDE + 8 * half + 4) = q3;
    }
    __asm__ volatile("" ::: "memory");  // same-wave DS is in-order; pin order

    // GEMM2: Out'[m=pixel][n] = E @ Q', n = {c0,c1,c2,sumE}; 8 chained WMMAs
    v8f acc = {};
#pragma unroll
    for (int j = 0; j < 8; ++j) {
      const int q = 4 * j + 2 * half;
      v2f av;
      av.x = myE[q * ESTRIDE + col];        // pair at dword offsets {0, 20}:
      av.y = myE[(q + 1) * ESTRIDE + col];  //   fuses to ds_load_2addr_b32
      v2f bv; bv.x = bx[j]; bv.y = by[j];
      acc = wmma_f32_16x16x4(av, bv, acc);
    }
    __asm__ volatile("" ::: "memory");

    // Pack D (lane=n, vgpr=pixel) into LDS for coalesced output
    if (col < 4) {
#pragma unroll
      for (int r = 0; r < 8; ++r)
        myO[(r + 8 * half) * 4 + col] = acc[r];
    }
    __asm__ volatile("" ::: "memory");

    // 48 contiguous output floats per tile: out[p][c] = num_c / sumE
    {
      const int i = lane;
      const int p = i / 3, c = i - 3 * p;
      outb[(size_t)pbase * 3 + i] =
          myO[p * 4 + c] * FAST_RCP(myO[p * 4 + 3]);
      const int i2 = i + 32;
      if (i2 < 48) {
        const int p2 = i2 / 3, c2 = i2 - 3 * p2;
        outb[(size_t)pbase * 3 + i2] =
            myO[p2 * 4 + c2] * FAST_RCP(myO[p2 * 4 + 3]);
      }
    }
    __asm__ volatile("" ::: "memory");

    vcur = vnext;  // rotate pipeline
  }
}

extern "C" void kernel_launch(void* const* d_in, const int* in_sizes, int n_in,
                              void* d_out, int out_size, void* d_ws,
                              size_t ws_size, hipStream_t stream) {
  (void)in_sizes; (void)n_in; (void)out_size; (void)d_ws; (void)ws_size;
  const float* img  = (const float*)d_in[0];
  const float* pal  = (const float*)d_in[1];
  const float* temp = (const float*)d_in[2];
  float* out        = (float*)d_out;
  hipLaunchKernelGGL(palette_quant_wmma, dim3(GRID), dim3(256), 0, stream,
                     img, pal, temp, out);
}